// GraphLearningEncoder_80264348827935
// MI455X (gfx1250) — compile-verified
//
#include <hip/hip_runtime.h>
#include <hip/hip_bf16.h>

typedef float v2f __attribute__((ext_vector_type(2)));
typedef float v8f __attribute__((ext_vector_type(8)));

#define BS 64
#define CH 16
#define TT 20
#define NN 100
#define KK (CH * TT)   // 320
#define NTILE 7        // ceil(100/16)

// ---------------------------------------------------------------------------
// Kernel 1: fused TCM (grouped temporal conv k=3 + ReLU) + full tconv
// (C,T)->C as an f32 WMMA GEMM:  D[16 x 16nodes] = W[16 x 320] * Htcm[320 x 16nodes]
// One wave (32 lanes) per (batch, 16-node column tile).
// ---------------------------------------------------------------------------
__global__ __launch_bounds__(32)
void nodefeat_wmma_kernel(const float* __restrict__ x,
                          const float* __restrict__ tcm_w,
                          const float* __restrict__ tcm_b,
                          const float* __restrict__ tconv_w,
                          const float* __restrict__ tconv_b,
                          float* __restrict__ h_ws) {
  __shared__ float hs[KK * 16];       // B-operand tile [k=320][col=16], 20480 B
  const int ct     = blockIdx.x;      // column tile 0..6
  const int b      = blockIdx.y;      // batch
  const int lane   = threadIdx.x;     // wave32
  const int lane15 = lane & 15;
  const int hi     = lane >> 4;       // 0/1 -> which half-wave
  const int n0     = ct * 16;
  const int n      = n0 + lane15;     // node column this lane owns
  const bool valid = (n < NN);

  // Stage TCM(x)+ReLU into LDS: lane covers column lane15, channels [hi*8, hi*8+8).
  {
    const int i0 = hi * 8;
    for (int i = i0; i < i0 + 8; ++i) {
      const float w0 = tcm_w[i * 3 + 0], w1 = tcm_w[i * 3 + 1], w2 = tcm_w[i * 3 + 2];
      const float bb = tcm_b[i];
      const float* xp = x + ((size_t)(b * CH + i) * TT) * NN + (valid ? n : 0);
      float xm = 0.f, xc = xp[0], xn;
      #pragma unroll
      for (int t = 0; t < TT; ++t) {
        xn = (t < TT - 1) ? xp[(t + 1) * NN] : 0.f;   // 'same' zero padding in T
        float v = fmaf(w0, xm, fmaf(w1, xc, fmaf(w2, xn, bb)));
        v = fmaxf(v, 0.f);
        hs[(i * TT + t) * 16 + lane15] = valid ? v : 0.f;  // zero padded columns
        xm = xc; xc = xn;
      }
    }
  }
  __syncthreads();

  // 80 x V_WMMA_F32_16X16X4_F32 over K=320.
  // f32 A/B layout (ISA 7.12.2): lanes 0-15 carry K = {kb, kb+1}, lanes 16-31
  // carry K = {kb+2, kb+3}; A lane index = M, B lane index = N.
  v8f acc = {0.f, 0.f, 0.f, 0.f, 0.f, 0.f, 0.f, 0.f};
  const float* wrow = tconv_w + lane15 * KK;   // A-matrix row M = lane15
  const int koff = hi ? 2 : 0;
  #pragma unroll 4
  for (int kk = 0; kk < KK / 4; ++kk) {
    const int ks = kk * 4 + koff;
    v2f a, bf;
    a.x  = wrow[ks];
    a.y  = wrow[ks + 1];
    bf.x = hs[ks * 16 + lane15];
    bf.y = hs[(ks + 1) * 16 + lane15];
    acc = __builtin_amdgcn_wmma_f32_16x16x4_f32(false, a, false, bf,
                                                (short)0, acc, false, false);
  }

  // D layout: VGPR r -> (m = r + hi*8, n = n0 + lane15).
  if (valid) {
    #pragma unroll
    for (int r = 0; r < 8; ++r) {
      const int m = r + hi * 8;
      h_ws[(size_t)(b * CH + m) * NN + n] = acc[r] + tconv_b[m];
    }
  }
}

// ---------------------------------------------------------------------------
// Kernel 2: per-(b,c) mean over nodes (edge2node collapses to a mean), then
// build the rank-structure vectors:
//   hn[n]  = (mean*w0 + h[n]*w1 + nb)*ew + eb
//   A[n]   = hn[n]*u0 + h[n]*w0*u2          (j-role / sender term)
//   B[n]   = hn[n]*u1 + h[n]*w1*u2 + nb*u2 + ub   (i-role / receiver term)
// out[i,j] = A[j] + B[i]
// ---------------------------------------------------------------------------
__global__ __launch_bounds__(32)
void edgevec_kernel(const float* __restrict__ h_ws,
                    const float* __restrict__ n2e_w, const float* __restrict__ n2e_b,
                    const float* __restrict__ e2n_w, const float* __restrict__ e2n_b,
                    const float* __restrict__ n2e2_w, const float* __restrict__ n2e2_b,
                    float* __restrict__ Avec, float* __restrict__ Bvec) {
  const int bc   = blockIdx.x;        // b*16 + c, 0..1023
  const int c    = bc & (CH - 1);
  const int lane = threadIdx.x;
  const float* h = h_ws + (size_t)bc * NN;

  float s = 0.f;
  for (int nn = lane; nn < NN; nn += 32) s += h[nn];
  #pragma unroll
  for (int off = 16; off > 0; off >>= 1) s += __shfl_xor(s, off, 32);
  const float mean = s * (1.0f / NN);

  const float w0 = n2e_w[c * 2 + 0], w1 = n2e_w[c * 2 + 1], nb = n2e_b[c];
  const float ew = e2n_w[c], eb = e2n_b[c];
  const float u0 = n2e2_w[c * 3 + 0], u1 = n2e2_w[c * 3 + 1], u2 = n2e2_w[c * 3 + 2];
  const float cst = n2e_b[c] * u2 + n2e2_b[c];

  for (int nn = lane; nn < NN; nn += 32) {
    const float hv = h[nn];
    const float hn = (mean * w0 + hv * w1 + nb) * ew + eb;
    Avec[(size_t)bc * NN + nn] = hn * u0 + hv * (w0 * u2);
    Bvec[(size_t)bc * NN + nn] = hn * u1 + hv * (w1 * u2) + cst;
  }
}

// ---------------------------------------------------------------------------
// Kernel 3: the only heavy kernel — stream 41 MB of out[b,c,i,j] = A[j]+B[i].
// One float4 per thread; rows are 400 B so every float4 is 16 B aligned.
// ---------------------------------------------------------------------------
__global__ __launch_bounds__(256)
void edge_write_kernel(const float* __restrict__ Avec,
                       const float* __restrict__ Bvec,
                       float* __restrict__ out) {
  const unsigned idx = blockIdx.x * 256u + threadIdx.x;  // 2,560,000 float4s total
  const unsigned row = idx / 25u;                        // (b*16+c)*100 + i
  const unsigned jf  = idx - row * 25u;                  // float4 index along j
  const unsigned bc  = row / 100u;
  const float    bi  = Bvec[row];
  const float4   a4  = *reinterpret_cast<const float4*>(Avec + bc * 100u + jf * 4u);
  float4 o;
  o.x = a4.x + bi; o.y = a4.y + bi; o.z = a4.z + bi; o.w = a4.w + bi;
  *reinterpret_cast<float4*>(out + (size_t)row * 100u + jf * 4u) = o;
}

extern "C" void kernel_launch(void* const* d_in, const int* in_sizes, int n_in,
                              void* d_out, int out_size, void* d_ws, size_t ws_size,
                              hipStream_t stream) {
  const float* x       = (const float*)d_in[0];
  // d_in[1] rel_rec, d_in[2] rel_send: structure handled analytically, unused.
  const float* tcm_w   = (const float*)d_in[3];
  const float* tcm_b   = (const float*)d_in[4];
  const float* tconv_w = (const float*)d_in[5];
  const float* tconv_b = (const float*)d_in[6];
  const float* n2e_w   = (const float*)d_in[7];
  const float* n2e_b   = (const float*)d_in[8];
  const float* e2n_w   = (const float*)d_in[9];
  const float* e2n_b   = (const float*)d_in[10];
  const float* n2e2_w  = (const float*)d_in[11];
  const float* n2e2_b  = (const float*)d_in[12];
  float* out = (float*)d_out;

  float* h_ws = (float*)d_ws;            // [64*16*100] = 409,600 B
  float* Avec = h_ws + BS * CH * NN;     // +409,600 B
  float* Bvec = Avec + BS * CH * NN;     // +409,600 B  (total 1.2 MB scratch)

  nodefeat_wmma_kernel<<<dim3(NTILE, BS), 32, 0, stream>>>(
      x, tcm_w, tcm_b, tconv_w, tconv_b, h_ws);
  edgevec_kernel<<<dim3(BS * CH), 32, 0, stream>>>(
      h_ws, n2e_w, n2e_b, e2n_w, e2n_b, n2e2_w, n2e2_b, Avec, Bvec);
  edge_write_kernel<<<dim3((BS * CH * NN * NN / 4) / 256), 256, 0, stream>>>(
      Avec, Bvec, out);
}